// MySCNN_44684839748190
// MI455X (gfx1250) — compile-verified
//
#include <hip/hip_runtime.h>

typedef float v2f __attribute__((ext_vector_type(2)));
typedef float v8f __attribute__((ext_vector_type(8)));

#define KT 32

// ---------------------------------------------------------------------------
// Layer-1 simplicial conv:  Z = L @ X  (M x 32), then
// H[(b*16+o), m] = leaky( sum_i th[o,i,0]*X[m,bi] + th[o,i,1]*Z[m,bi] + bias[o] )
// H stored transposed (128, M) row-major so layer 2 reads contiguous K.
// ---------------------------------------------------------------------------
__global__ __launch_bounds__(128) void sconv1_kernel(
    const float* __restrict__ L, const float* __restrict__ x,
    const float* __restrict__ th, const float* __restrict__ bias,
    float* __restrict__ H, int M)
{
    __shared__ float Llds[32][36];     // stride 36: float4-aligned, conflict-free A reads
    __shared__ float Xlds[KT][33];
    __shared__ float Zlds[32][33];
    __shared__ float Xe[32][33];

    const int tid  = threadIdx.x;
    const int lane = tid & 31;
    const int wave = tid >> 5;
    const int ln   = lane & 15;
    const int half = lane >> 4;
    const int r    = wave >> 1;   // row tile within block (0/1)
    const int c    = wave & 1;    // col tile (0/1)
    const int m0   = blockIdx.x * 32;
    const bool fullrows = (m0 + 32 <= M);

    // Stage x values at the output rows (needed by the theta epilogue).
    if (fullrows) {
#pragma unroll
        for (int i = 0; i < 2; ++i) {
            int f = tid + i * 128;
            int col = f >> 3, j4 = (f & 7) * 4;
            float4 v = *(const float4*)&x[(long)col * M + m0 + j4];
            Xe[j4 + 0][col] = v.x; Xe[j4 + 1][col] = v.y;
            Xe[j4 + 2][col] = v.z; Xe[j4 + 3][col] = v.w;
        }
    } else {
        for (int e = tid; e < 32 * 32; e += 128) {
            int col = e >> 5, j = e & 31;
            int m = m0 + j;
            Xe[j][col] = (m < M) ? x[(long)col * M + m] : 0.f;
        }
    }

    v8f acc = {0.f, 0.f, 0.f, 0.f, 0.f, 0.f, 0.f, 0.f};

    for (int k0 = 0; k0 < M; k0 += KT) {
        __syncthreads();
        if (fullrows && (k0 + KT <= M)) {
            // Fast path: vector loads, no predication.
            {
                int rbase = tid >> 3, k4 = (tid & 7) * 4;
#pragma unroll
                for (int i = 0; i < 2; ++i) {
                    int row = rbase + i * 16;
                    float4 v = *(const float4*)&L[(long)(m0 + row) * M + k0 + k4];
                    *(float4*)&Llds[row][k4] = v;
                }
            }
#pragma unroll
            for (int i = 0; i < 2; ++i) {
                int f = tid + i * 128;
                int col = f >> 3, k4 = (f & 7) * 4;
                float4 v = *(const float4*)&x[(long)col * M + k0 + k4];
                Xlds[k4 + 0][col] = v.x; Xlds[k4 + 1][col] = v.y;
                Xlds[k4 + 2][col] = v.z; Xlds[k4 + 3][col] = v.w;
            }
        } else {
            for (int e = tid; e < 32 * KT; e += 128) {
                int rr = e >> 5, kk = e & 31;
                int row = m0 + rr, k = k0 + kk;
                if (row >= M) row = M - 1;           // clamp: garbage rows never read
                Llds[rr][kk] = (k < M) ? L[(long)row * M + k] : 0.f;
            }
            for (int e = tid; e < 32 * KT; e += 128) {
                int col = e >> 5, kk = e & 31;
                int k = k0 + kk;
                Xlds[kk][col] = (k < M) ? x[(long)col * M + k] : 0.f;
            }
        }
        __syncthreads();
#pragma unroll
        for (int kk = 0; kk < KT; kk += 4) {
            v2f a, b;
            a.x = Llds[r * 16 + ln][kk + half * 2 + 0];
            a.y = Llds[r * 16 + ln][kk + half * 2 + 1];
            b.x = Xlds[kk + half * 2 + 0][c * 16 + ln];
            b.y = Xlds[kk + half * 2 + 1][c * 16 + ln];
            acc = __builtin_amdgcn_wmma_f32_16x16x4_f32(
                false, a, false, b, (short)0, acc, false, false);
        }
    }

    __syncthreads();
#pragma unroll
    for (int v = 0; v < 8; ++v)
        Zlds[r * 16 + v + 8 * half][c * 16 + ln] = acc[v];
    __syncthreads();

    // Epilogue: thread t owns output column (b, o) with b = t>>4, o = t&15.
    const int b_ = tid >> 4, o = tid & 15;
    float t0[4], t1[4];
#pragma unroll
    for (int i = 0; i < 4; ++i) {
        t0[i] = th[o * 8 + i * 2 + 0];
        t1[i] = th[o * 8 + i * 2 + 1];
    }
    const float bs = bias[o];
    for (int rr = 0; rr < 32; ++rr) {
        int m = m0 + rr;
        if (m >= M) break;
        float v = bs;
#pragma unroll
        for (int i = 0; i < 4; ++i)
            v += t0[i] * Xe[rr][b_ * 4 + i] + t1[i] * Zlds[rr][b_ * 4 + i];
        H[(long)tid * M + m] = (v >= 0.f) ? v : 0.01f * v;
    }
}

// ---------------------------------------------------------------------------
// Layer-2 simplicial conv:  Z2 = L @ H  (M x 128), then
// out[b,o,m] = sum_i th2[o,i,0]*H[m,(b,i)] + th2[o,i,1]*Z2[m,(b,i)]
// Also accumulates column sums (for mean pooling) into pool[b*32+o].
// ---------------------------------------------------------------------------
__global__ __launch_bounds__(128) void sconv2_kernel(
    const float* __restrict__ L, const float* __restrict__ H,
    const float* __restrict__ th2, float* __restrict__ out,
    float* __restrict__ pool, int M)
{
    __shared__ float Llds[16][36];
    __shared__ float Hlds[KT][132];
    __shared__ float Zlds[16][132];
    __shared__ float He[16][132];

    const int tid  = threadIdx.x;
    const int lane = tid & 31;
    const int wave = tid >> 5;
    const int ln   = lane & 15;
    const int half = lane >> 4;
    const int m0   = blockIdx.x * 16;
    const bool fullrows = (m0 + 16 <= M);

    // Stage H at output rows for the epilogue.
    if (fullrows) {
#pragma unroll
        for (int i = 0; i < 4; ++i) {
            int f = tid + i * 128;
            int col = f >> 2, j4 = (f & 3) * 4;
            float4 v = *(const float4*)&H[(long)col * M + m0 + j4];
            He[j4 + 0][col] = v.x; He[j4 + 1][col] = v.y;
            He[j4 + 2][col] = v.z; He[j4 + 3][col] = v.w;
        }
    } else {
        for (int e = tid; e < 16 * 128; e += 128) {
            int col = e >> 4, j = e & 15;
            int m = m0 + j;
            He[j][col] = (m < M) ? H[(long)col * M + m] : 0.f;
        }
    }

    v8f acc0 = {0.f, 0.f, 0.f, 0.f, 0.f, 0.f, 0.f, 0.f};
    v8f acc1 = {0.f, 0.f, 0.f, 0.f, 0.f, 0.f, 0.f, 0.f};

    for (int k0 = 0; k0 < M; k0 += KT) {
        __syncthreads();
        if (fullrows && (k0 + KT <= M)) {
            {
                int row = tid >> 3, k4 = (tid & 7) * 4;
                float4 v = *(const float4*)&L[(long)(m0 + row) * M + k0 + k4];
                *(float4*)&Llds[row][k4] = v;
            }
#pragma unroll
            for (int i = 0; i < 8; ++i) {
                int f = tid + i * 128;
                int col = f >> 3, k4 = (f & 7) * 4;
                float4 v = *(const float4*)&H[(long)col * M + k0 + k4];
                Hlds[k4 + 0][col] = v.x; Hlds[k4 + 1][col] = v.y;
                Hlds[k4 + 2][col] = v.z; Hlds[k4 + 3][col] = v.w;
            }
        } else {
            for (int e = tid; e < 16 * KT; e += 128) {
                int rr = e >> 5, kk = e & 31;
                int row = m0 + rr, k = k0 + kk;
                if (row >= M) row = M - 1;
                Llds[rr][kk] = (k < M) ? L[(long)row * M + k] : 0.f;
            }
            for (int e = tid; e < 128 * KT; e += 128) {
                int col = e >> 5, kk = e & 31;
                int k = k0 + kk;
                Hlds[kk][col] = (k < M) ? H[(long)col * M + k] : 0.f;
            }
        }
        __syncthreads();
#pragma unroll
        for (int kk = 0; kk < KT; kk += 4) {
            v2f a, b0, b1;
            a.x = Llds[ln][kk + half * 2 + 0];
            a.y = Llds[ln][kk + half * 2 + 1];
            b0.x = Hlds[kk + half * 2 + 0][wave * 32 + ln];
            b0.y = Hlds[kk + half * 2 + 1][wave * 32 + ln];
            b1.x = Hlds[kk + half * 2 + 0][wave * 32 + 16 + ln];
            b1.y = Hlds[kk + half * 2 + 1][wave * 32 + 16 + ln];
            acc0 = __builtin_amdgcn_wmma_f32_16x16x4_f32(
                false, a, false, b0, (short)0, acc0, false, false);
            acc1 = __builtin_amdgcn_wmma_f32_16x16x4_f32(
                false, a, false, b1, (short)0, acc1, false, false);
        }
    }

    __syncthreads();
#pragma unroll
    for (int v = 0; v < 8; ++v) {
        Zlds[v + 8 * half][wave * 32 + ln]      = acc0[v];
        Zlds[v + 8 * half][wave * 32 + 16 + ln] = acc1[v];
    }
    __syncthreads();

    // Epilogue: 256 output columns (b*32+o), 2 per thread.
    for (int cc = tid; cc < 256; cc += 128) {
        const int b_ = cc >> 5, o = cc & 31;
        float c0[16], c1[16];
#pragma unroll
        for (int i = 0; i < 16; ++i) {
            c0[i] = th2[o * 32 + i * 2 + 0];
            c1[i] = th2[o * 32 + i * 2 + 1];
        }
        float colsum = 0.f;
        for (int rr = 0; rr < 16; ++rr) {
            int m = m0 + rr;
            if (m >= M) break;
            float v = 0.f;
#pragma unroll
            for (int i = 0; i < 16; ++i)
                v += c0[i] * He[rr][b_ * 16 + i] + c1[i] * Zlds[rr][b_ * 16 + i];
            out[((long)b_ * 32 + o) * M + m] = v;
            colsum += v;
        }
        atomicAdd(&pool[cc], colsum);
    }
}

// ---------------------------------------------------------------------------
// Pool-normalize + 4 linear layers (reference MLP has no activations).
// ---------------------------------------------------------------------------
__global__ __launch_bounds__(768) void mlp_kernel(
    const float* __restrict__ pool,
    const float* __restrict__ W1, const float* __restrict__ bb1,
    const float* __restrict__ W2, const float* __restrict__ bb2,
    const float* __restrict__ W3, const float* __restrict__ bb3,
    const float* __restrict__ W4, const float* __restrict__ bb4,
    float* __restrict__ logits, float inv0, float inv1, float inv2)
{
    __shared__ float P[8][96];
    __shared__ float h1[8][64];
    __shared__ float h2[8][32];
    __shared__ float h3[8][16];
    const int t = threadIdx.x;
    if (t < 768) {
        int b = t / 96, k = t % 96;
        int d = k / 32, o = k % 32;
        float inv = (d == 0) ? inv0 : ((d == 1) ? inv1 : inv2);
        P[b][k] = pool[d * 256 + b * 32 + o] * inv;
    }
    __syncthreads();
    if (t < 512) {
        int b = t >> 6, j = t & 63;
        float v = bb1[j];
        for (int k = 0; k < 96; ++k) v += P[b][k] * W1[j * 96 + k];
        h1[b][j] = v;
    }
    __syncthreads();
    if (t < 256) {
        int b = t >> 5, j = t & 31;
        float v = bb2[j];
        for (int k = 0; k < 64; ++k) v += h1[b][k] * W2[j * 64 + k];
        h2[b][j] = v;
    }
    __syncthreads();
    if (t < 128) {
        int b = t >> 4, j = t & 15;
        float v = bb3[j];
        for (int k = 0; k < 32; ++k) v += h2[b][k] * W3[j * 32 + k];
        h3[b][j] = v;
    }
    __syncthreads();
    if (t < 16) {
        int b = t >> 1, j = t & 1;
        float v = bb4[j];
        for (int k = 0; k < 16; ++k) v += h3[b][k] * W4[j * 16 + k];
        logits[b * 2 + j] = v;
    }
}

__global__ void zero_kernel(float* p, int n)
{
    int i = blockIdx.x * blockDim.x + threadIdx.x;
    if (i < n) p[i] = 0.f;
}

// ---------------------------------------------------------------------------
extern "C" void kernel_launch(void* const* d_in, const int* in_sizes, int n_in,
                              void* d_out, int out_size, void* d_ws, size_t ws_size,
                              hipStream_t stream)
{
    const float* Lm[3]  = {(const float*)d_in[0], (const float*)d_in[5],  (const float*)d_in[10]};
    const float* xm[3]  = {(const float*)d_in[1], (const float*)d_in[6],  (const float*)d_in[11]};
    const float* th1[3] = {(const float*)d_in[2], (const float*)d_in[7],  (const float*)d_in[12]};
    const float* bi1[3] = {(const float*)d_in[3], (const float*)d_in[8],  (const float*)d_in[13]};
    const float* th2[3] = {(const float*)d_in[4], (const float*)d_in[9],  (const float*)d_in[14]};
    const float* W1  = (const float*)d_in[15]; const float* bb1 = (const float*)d_in[16];
    const float* W2  = (const float*)d_in[17]; const float* bb2 = (const float*)d_in[18];
    const float* W3  = (const float*)d_in[19]; const float* bb3 = (const float*)d_in[20];
    const float* W4  = (const float*)d_in[21]; const float* bb4 = (const float*)d_in[22];

    int Ms[3];
    for (int d = 0; d < 3; ++d) Ms[d] = in_sizes[1 + 5 * d] / 32;  // x is (8,4,M)

    float* ws   = (float*)d_ws;
    float* pool = ws;            // 3 * 256 floats of pooled column sums
    float* Hbuf[3];
    size_t off = 1024;
    for (int d = 0; d < 3; ++d) { Hbuf[d] = ws + off; off += (size_t)128 * Ms[d]; }

    float* outp   = (float*)d_out;
    float* logits = outp;        // first 16 floats
    float* out2[3];
    out2[0] = outp + 16;
    out2[1] = out2[0] + (size_t)8 * 32 * Ms[0];
    out2[2] = out2[1] + (size_t)8 * 32 * Ms[1];

    zero_kernel<<<4, 256, 0, stream>>>(pool, 768);

    for (int d = 0; d < 3; ++d)
        sconv1_kernel<<<(Ms[d] + 31) / 32, 128, 0, stream>>>(
            Lm[d], xm[d], th1[d], bi1[d], Hbuf[d], Ms[d]);

    for (int d = 0; d < 3; ++d)
        sconv2_kernel<<<(Ms[d] + 15) / 16, 128, 0, stream>>>(
            Lm[d], Hbuf[d], th2[d], out2[d], pool + d * 256, Ms[d]);

    mlp_kernel<<<1, 768, 0, stream>>>(pool, W1, bb1, W2, bb2, W3, bb3, W4, bb4,
                                      logits, 1.f / Ms[0], 1.f / Ms[1], 1.f / Ms[2]);
}